// ZFPrecoder_64828236366368
// MI455X (gfx1250) — compile-verified
//
#include <hip/hip_runtime.h>
#include <cstdint>

// ZF precoder: per (b, ofdm, fft): H (4x16 cpx), A = H H^H, u = A^{-1} x, y = H^H u.
// B=16, S=4, T=16 (NRXA*T), OFDM=14, FFT=1024, NTX=1.
// H tile staged into LDS via the Tensor Data Mover (CDNA5 TDM), reused twice on-chip.

#define TPB 128

static constexpr int kB    = 16;
static constexpr int kS    = 4;
static constexpr int kT    = 16;
static constexpr int kOFDM = 14;
static constexpr int kFFT  = 1024;
static constexpr int kST   = kS * kT;          // 64 rows in the H tile
static constexpr int kChunks = kFFT / TPB;     // 8
static constexpr int kHRow = kOFDM * kFFT;     // 14336 dwords: global row stride over t

typedef uint32_t v4u __attribute__((ext_vector_type(4)));
typedef int      v8i __attribute__((ext_vector_type(8)));
typedef int      v4i __attribute__((ext_vector_type(4)));

struct cpx { float re, im; };

__device__ __forceinline__ cpx cadd(cpx a, cpx b) { return {a.re + b.re, a.im + b.im}; }
__device__ __forceinline__ cpx csub(cpx a, cpx b) { return {a.re - b.re, a.im - b.im}; }
__device__ __forceinline__ cpx cmul(cpx a, cpx b) {           // a * b
    return {fmaf(a.re, b.re, -(a.im * b.im)), fmaf(a.re, b.im, a.im * b.re)};
}
__device__ __forceinline__ cpx cmulc(cpx a, cpx b) {          // a * conj(b)
    return {fmaf(a.re, b.re, a.im * b.im), fmaf(a.im, b.re, -(a.re * b.im))};
}
__device__ __forceinline__ cpx cjmul(cpx a, cpx b) {          // conj(a) * b
    return {fmaf(a.re, b.re, a.im * b.im), fmaf(a.re, b.im, -(a.im * b.re))};
}
__device__ __forceinline__ cpx cconj(cpx a)           { return {a.re, -a.im}; }
__device__ __forceinline__ cpx cscale(cpx a, float s) { return {a.re * s, a.im * s}; }
__device__ __forceinline__ float cnorm(cpx a) { return fmaf(a.re, a.re, a.im * a.im); }

__global__ __launch_bounds__(TPB) void zf_precoder_kernel(
    const float* __restrict__ xr, const float* __restrict__ xi,
    const float* __restrict__ hr, const float* __restrict__ hi,
    const int*   __restrict__ pind,
    float2* __restrict__ out)
{
    // H tile for this block: [re/im][s*16+t][fft_in_tile]  -> 64 KB
    __shared__ __align__(16) float lds_h[2][kST][TPB];

    const int tid   = threadIdx.x;
    const int chunk = blockIdx.x & (kChunks - 1);
    const int bo    = blockIdx.x >> 3;               // kChunks == 8
    const int o     = bo % kOFDM;
    const int b     = bo / kOFDM;
    const int f0    = chunk * TPB;
    const int f     = f0 + tid;

    // precoding indices (identity in the reference harness, but honor them)
    const int4 pv = *reinterpret_cast<const int4*>(pind);

    // ---------------- x vector: issue early, overlaps the tensor DMA -------
    const size_t xrow = ((size_t)(b * 4) * kOFDM + o) * kFFT + f;
    const size_t xstr = (size_t)kOFDM * kFFT;
    const cpx x0 = { xr[xrow           ], xi[xrow           ] };
    const cpx x1 = { xr[xrow +     xstr], xi[xrow +     xstr] };
    const cpx x2 = { xr[xrow + 2 * xstr], xi[xrow + 2 * xstr] };
    const cpx x3 = { xr[xrow + 3 * xstr], xi[xrow + 3 * xstr] };

    // ---------------- TDM stage: global -> LDS ------------------------------
    // Per (plane, s): 2D tile, 16 rows (t) x 128 dwords (fft), global row
    // stride 14336 dwords, contiguous in LDS. One tensor_load_to_lds each.
    // TDM ignores EXEC; each of the 4 waves issues 2 of the 8 descriptors so
    // the DMA engine can overlap them (same-wave tensor ops are ordered,
    // cross-wave ops are not).
    const uint32_t lds_base = (uint32_t)(uintptr_t)(&lds_h[0][0][0]);
    {
        // D# group 1 (256b): workgroup_mask=0, data_size=4B (2), no barrier/
        // iterate/pad; tensor_dim0=128 [bits79:48], tensor_dim1=16 [111:80],
        // tile_dim0=128 [127:112], tile_dim1=16 [143:128], tile_dim2=0,
        // tensor_dim0_stride=14336 [207:160], tensor_dim1_stride=0.
        const v8i g1 = { (int)0x00020000,            // data_size=2 (4 bytes)
                         (int)(128u << 16),          // tensor_dim0[15:0]
                         (int)(16u  << 16),          // dim0[31:16]=0 | tensor_dim1[15:0]
                         (int)(128u << 16),          // dim1[31:16]=0 | tile_dim0=128
                         (int)16,                    // tile_dim1=16 | tile_dim2=0
                         (int)kHRow,                 // tensor_dim0_stride[31:0]=14336
                         (int)0,                     // stride[47:32]=0 | dim1_stride[15:0]=0
                         (int)0 };                   // tensor_dim1_stride[47:16]=0
        const v4i gz  = { 0, 0, 0, 0 };              // groups 2/3: 2-D tensor
        const v8i gz8 = { 0, 0, 0, 0, 0, 0, 0, 0 };  // trailing group (unused, zero)
        const int wave = tid >> 5;                   // 0..3, wave-uniform
        #pragma unroll
        for (int k = 0; k < 2; ++k) {
            const int idx   = wave * 2 + k;          // 0..7 = plane*4 + s
            const int plane = idx >> 2;
            const int s     = idx & 3;
            const int r = (s < 2) ? (s == 0 ? pv.x : pv.y)
                                  : (s == 2 ? pv.z : pv.w);
            const float* src = plane ? hi : hr;
            const uint64_t gaddr = (uint64_t)(uintptr_t)(src +
                ((size_t)((b * 4 + r) * 16) * (size_t)kHRow
                 + (size_t)o * kFFT + f0));
            const uint32_t laddr = lds_base +
                (uint32_t)(plane * kST + s * kT) * (TPB * 4u);
            // D# group 0: count=1 | lds_addr | global_addr[56:0] | type=2
            const v4u g0 = { 1u,
                             laddr,
                             (uint32_t)gaddr,
                             (uint32_t)(gaddr >> 32) | 0x80000000u };
            __builtin_amdgcn_tensor_load_to_lds(g0, g1, gz, gz, gz8, 0);
        }
        __builtin_amdgcn_s_wait_tensorcnt(0);
    }
    __syncthreads();   // every wave drained its own TENSORcnt before arriving

    // ---------------- pass 1: A = H H^H (Hermitian: 4 real diag + 6 upper) --
    float d0 = 0.f, d1 = 0.f, d2 = 0.f, d3 = 0.f;
    cpx a01{0,0}, a02{0,0}, a03{0,0}, a12{0,0}, a13{0,0}, a23{0,0};
    #pragma unroll
    for (int t = 0; t < kT; ++t) {
        const cpx h0 = { lds_h[0][ 0 + t][tid], lds_h[1][ 0 + t][tid] };
        const cpx h1 = { lds_h[0][16 + t][tid], lds_h[1][16 + t][tid] };
        const cpx h2 = { lds_h[0][32 + t][tid], lds_h[1][32 + t][tid] };
        const cpx h3 = { lds_h[0][48 + t][tid], lds_h[1][48 + t][tid] };
        d0 += cnorm(h0); d1 += cnorm(h1); d2 += cnorm(h2); d3 += cnorm(h3);
        a01 = cadd(a01, cmulc(h0, h1));
        a02 = cadd(a02, cmulc(h0, h2));
        a03 = cadd(a03, cmulc(h0, h3));
        a12 = cadd(a12, cmulc(h1, h2));
        a13 = cadd(a13, cmulc(h1, h3));
        a23 = cadd(a23, cmulc(h2, h3));
    }

    // ---------------- LDL^H factorization of A (A[i][j], i>j = conj(a_ji)) --
    const float iD0 = 1.0f / d0;
    const cpx L10 = cscale(cconj(a01), iD0);
    const cpx L20 = cscale(cconj(a02), iD0);
    const cpx L30 = cscale(cconj(a03), iD0);
    const float D1  = d1 - cnorm(L10) * d0;
    const float iD1 = 1.0f / D1;
    const cpx L21 = cscale(csub(cconj(a12), cscale(cmulc(L20, L10), d0)), iD1);
    const cpx L31 = cscale(csub(cconj(a13), cscale(cmulc(L30, L10), d0)), iD1);
    const float D2  = d2 - cnorm(L20) * d0 - cnorm(L21) * D1;
    const float iD2 = 1.0f / D2;
    const cpx L32 = cscale(csub(csub(cconj(a23), cscale(cmulc(L30, L20), d0)),
                                cscale(cmulc(L31, L21), D1)), iD2);
    const float D3  = d3 - cnorm(L30) * d0 - cnorm(L31) * D1 - cnorm(L32) * D2;
    const float iD3 = 1.0f / D3;

    // ---------------- solve A u = x ---------------------------------------
    const cpx z0 = x0;
    const cpx z1 = csub(x1, cmul(L10, z0));
    const cpx z2 = csub(csub(x2, cmul(L20, z0)), cmul(L21, z1));
    const cpx z3 = csub(csub(csub(x3, cmul(L30, z0)), cmul(L31, z1)), cmul(L32, z2));
    const cpx w0 = cscale(z0, iD0), w1 = cscale(z1, iD1);
    const cpx w2 = cscale(z2, iD2), w3 = cscale(z3, iD3);
    const cpx u3 = w3;
    const cpx u2 = csub(w2, cjmul(L32, u3));
    const cpx u1 = csub(csub(w1, cjmul(L21, u2)), cjmul(L31, u3));
    const cpx u0 = csub(csub(csub(w0, cjmul(L10, u1)), cjmul(L20, u2)), cjmul(L30, u3));

    // ---------------- pass 2: y[t] = sum_s conj(H[s][t]) u[s]  -------------
    #pragma unroll
    for (int t = 0; t < kT; ++t) {
        const cpx h0 = { lds_h[0][ 0 + t][tid], lds_h[1][ 0 + t][tid] };
        const cpx h1 = { lds_h[0][16 + t][tid], lds_h[1][16 + t][tid] };
        const cpx h2 = { lds_h[0][32 + t][tid], lds_h[1][32 + t][tid] };
        const cpx h3 = { lds_h[0][48 + t][tid], lds_h[1][48 + t][tid] };
        cpx y = cmulc(u0, h0);
        y = cadd(y, cmulc(u1, h1));
        y = cadd(y, cmulc(u2, h2));
        y = cadd(y, cmulc(u3, h3));
        out[((size_t)(b * 16 + t) * kOFDM + o) * kFFT + f] = make_float2(y.re, y.im);
    }
}

extern "C" void kernel_launch(void* const* d_in, const int* in_sizes, int n_in,
                              void* d_out, int out_size, void* d_ws, size_t ws_size,
                              hipStream_t stream) {
    (void)in_sizes; (void)n_in; (void)d_ws; (void)ws_size; (void)out_size;
    const float* xr   = (const float*)d_in[0];
    const float* xi   = (const float*)d_in[1];
    const float* hr   = (const float*)d_in[2];
    const float* hi   = (const float*)d_in[3];
    const int*   pind = (const int*)  d_in[4];
    float2* out = (float2*)d_out;   // complex64 -> interleaved (re, im) floats

    dim3 grid(kB * kOFDM * kChunks);   // 16 * 14 * 8 = 1792 blocks
    dim3 block(TPB);                   // 128 threads = 4 waves (wave32)
    zf_precoder_kernel<<<grid, block, 0, stream>>>(xr, xi, hr, hi, pind, out);
}